// ARNOLD_ReceptiveFieldEncoder_52639119180423
// MI455X (gfx1250) — compile-verified
//
#include <hip/hip_runtime.h>
#include <hip/hip_bf16.h>

typedef __attribute__((ext_vector_type(4))) float v4f;

// Fast path: N == 16. One thread per batch element b.
// Thread b owns out[t, b, 0..15] for all t: 16 contiguous floats (64B) per
// time plane -> adjacent lanes write adjacent 64B rows -> wave stores are
// fully contiguous. 4x nontemporal b128 stores per plane (output is 512MiB,
// 2.7x the 192MB L2, so stream past the cache).
__global__ __launch_bounds__(256) void rf_encode_n16(
    const float* __restrict__ x,
    const float* __restrict__ center,
    const float* __restrict__ scaling,
    float* __restrict__ out,
    int B, int T) {
  const int b = blockIdx.x * 256 + threadIdx.x;
  if (b >= B) return;

  const float xv = x[b];

  // Spike times for this sample across all 16 fields (registers).
  // center/scaling are wave-uniform -> scalar loads + broadcast.
  int ts[16];
#pragma unroll
  for (int n = 0; n < 16; ++n) {
    const float d = fabsf(xv - center[n]);
    int t = (int)(scaling[n] * d);   // trunc toward zero == reference .int()
    t = (t < 0) ? 0 : t;             // defensive (product is >= 0)
    ts[n] = (t > T - 1) ? (T - 1) : t;
  }

  const size_t plane = (size_t)B * 16;          // elements per time plane
  float* row = out + (size_t)b * 16;            // out[t=0, b, 0]

  for (int t = 0; t < T; ++t, row += plane) {
#pragma unroll
    for (int q = 0; q < 4; ++q) {
      v4f v;
      v.x = (ts[q * 4 + 0] == t) ? 1.0f : 0.0f;
      v.y = (ts[q * 4 + 1] == t) ? 1.0f : 0.0f;
      v.z = (ts[q * 4 + 2] == t) ? 1.0f : 0.0f;
      v.w = (ts[q * 4 + 3] == t) ? 1.0f : 0.0f;
      __builtin_nontemporal_store(v, (v4f*)(row + q * 4));
    }
  }
}

// Generic fallback for arbitrary N: one thread per output element,
// contiguous scalar NT stores (still fully coalesced).
__global__ __launch_bounds__(256) void rf_encode_generic(
    const float* __restrict__ x,
    const float* __restrict__ center,
    const float* __restrict__ scaling,
    float* __restrict__ out,
    long long total, int B, int N, int T) {
  long long i = (long long)blockIdx.x * 256 + threadIdx.x;
  if (i >= total) return;

  const int n = (int)(i % N);
  const long long r = i / N;
  const int b = (int)(r % B);
  const int t = (int)(r / B);

  int tsp = (int)(scaling[n] * fabsf(x[b] - center[n]));
  tsp = (tsp < 0) ? 0 : tsp;
  tsp = (tsp > T - 1) ? (T - 1) : tsp;

  __builtin_nontemporal_store((tsp == t) ? 1.0f : 0.0f, out + i);
}

extern "C" void kernel_launch(void* const* d_in, const int* in_sizes, int n_in,
                              void* d_out, int out_size, void* d_ws, size_t ws_size,
                              hipStream_t stream) {
  const float* x       = (const float*)d_in[0];
  const float* center  = (const float*)d_in[1];
  const float* scaling = (const float*)d_in[2];
  // d_in[3] is time_steps on device; recover T from sizes instead (host-side,
  // graph-capture safe): out_size = T * B * N.
  const int B = in_sizes[0];
  const int N = in_sizes[1];
  float* out = (float*)d_out;

  const long long total = (long long)out_size;
  const int T = (int)(total / ((long long)B * N));

  if (N == 16) {
    const int grid = (B + 255) / 256;
    rf_encode_n16<<<grid, 256, 0, stream>>>(x, center, scaling, out, B, T);
  } else {
    const long long grid = (total + 255) / 256;
    rf_encode_generic<<<(int)grid, 256, 0, stream>>>(x, center, scaling, out,
                                                     total, B, N, T);
  }
}